// QuantSoftmax_7687991459867
// MI455X (gfx1250) — compile-verified
//
#include <hip/hip_runtime.h>
#include <cstdint>
#include <cstddef>

// ---- I-BERT constants (pre-divided like the reference) ----
#define X0F        (-0.6931f)
#define COEF_A     0.35815147f
#define COEF_B_A   (0.96963238f / 0.35815147f)
#define COEF_C_A   (1.0f / 0.35815147f)

typedef __attribute__((ext_vector_type(2))) float v2f;
typedef __attribute__((ext_vector_type(4))) float v4f;   // native vector for NT stores
typedef __attribute__((ext_vector_type(8))) float v8f;

// ---------------- pass 0: zero the atomic accumulator ----------------
__global__ void init_ws_kernel(unsigned int* __restrict__ w) {
    if (threadIdx.x == 0) w[0] = 0u;
}

// ---------------- pass A: global abs-max ----------------
// scale = max(|min(x,0)|, |max(x,0)|)/n == max|x|/n; non-negative floats are
// order-isomorphic to their uint bit patterns, so one atomicMax(uint) per
// block finishes the reduction. Loads stay RT so the tensor is L2-resident
// for pass B (fits: 100 MB < 192 MB).
__global__ __launch_bounds__(256) void absmax_kernel(const float* __restrict__ x,
                                                     unsigned int* __restrict__ absbits,
                                                     int n4) {
    const float4* __restrict__ x4 = (const float4*)x;
    const int stride = gridDim.x * blockDim.x;
    float m = 0.0f;
    for (int i = blockIdx.x * blockDim.x + threadIdx.x; i < n4; i += stride) {
        __builtin_prefetch(&x4[i + 8 * stride], 0, 0);   // global_prefetch_b8
        float4 v = x4[i];
        m = fmaxf(m, fmaxf(fmaxf(fabsf(v.x), fabsf(v.y)),
                           fmaxf(fabsf(v.z), fabsf(v.w))));
    }
#pragma unroll
    for (int off = 16; off > 0; off >>= 1)
        m = fmaxf(m, __shfl_xor(m, off, 32));
    __shared__ float smax[8];
    const int wave = threadIdx.x >> 5;
    if ((threadIdx.x & 31) == 0) smax[wave] = m;
    __syncthreads();
    if (threadIdx.x == 0) {
        float mm = smax[0];
#pragma unroll
        for (int w = 1; w < 8; ++w) mm = fmaxf(mm, smax[w]);
        atomicMax(absbits, __float_as_uint(mm));
    }
}

// ---------------- pass B: fused int-softmax, one wave32 per 1024-row ----------------
__global__ __launch_bounds__(256) void isoftmax_kernel(const float* __restrict__ x,
                                                       float* __restrict__ out,
                                                       const unsigned int* __restrict__ absbits,
                                                       int n_rows) {
    const int lane = threadIdx.x & 31;
    const int wave = threadIdx.x >> 5;
    const int row  = blockIdx.x * 8 + wave;
    if (row >= n_rows) return;   // wave-uniform: EXEC all-ones inside live waves (WMMA-safe)

    // ---- uniform scalars (second global quantization folded analytically) ----
    const float amax      = __uint_as_float(absbits[0]);
    const float sf        = amax * (1.0f / 32767.0f);
    const float inv_sf    = 1.0f / sf;
    const float x0i       = floorf(X0F * inv_sf);                  // floor(X0/sf) < 0
    const float inv_x0i   = 1.0f / x0i;
    const float b_i       = floorf(COEF_B_A * inv_sf);
    const float c_i       = floorf(COEF_C_A * (inv_sf * inv_sf));
    const float exp_sf    = (COEF_A * sf * sf) * (1.0f / 1073741824.0f);   // /2^30
    const float emax      = floorf(c_i * 1073741824.0f) * exp_sf;  // max(exp_real), hit in every row
    const float exp_scale = emax * (1.0f / 32767.0f);
    const float s2        = exp_sf / exp_scale;                    // fused exp_real/exp_scale
    const float lo_clip   = 30.0f * x0i;                           // CONST * x0_int

    const float4* __restrict__ row4  = (const float4*)(x   + (size_t)row * 1024);
    v4f*          __restrict__ orow4 = (v4f*)         (out + (size_t)row * 1024);

    // ---- phase 1: quantize to integer-valued floats + row max (row in 32 VGPRs/lane) ----
    float v[32];
    float rmax = -3.402823466e38f;
#pragma unroll
    for (int k = 0; k < 8; ++k) {
        float4 t = row4[k * 32 + lane];                 // coalesced global_load_b128 (L2 hits)
        float a0 = fminf(fmaxf(rintf(t.x * inv_sf), -32768.0f), 32767.0f);  // v_med3
        float a1 = fminf(fmaxf(rintf(t.y * inv_sf), -32768.0f), 32767.0f);
        float a2 = fminf(fmaxf(rintf(t.z * inv_sf), -32768.0f), 32767.0f);
        float a3 = fminf(fmaxf(rintf(t.w * inv_sf), -32768.0f), 32767.0f);
        v[4*k+0] = a0; v[4*k+1] = a1; v[4*k+2] = a2; v[4*k+3] = a3;
        rmax = fmaxf(rmax, fmaxf(fmaxf(a0, a1), fmaxf(a2, a3)));
    }
#pragma unroll
    for (int off = 16; off > 0; off >>= 1)
        rmax = fmaxf(rmax, __shfl_xor(rmax, off, 32));

    // ---- phase 2: int exp poly + requant (exact 0 at argmax -> e2 = 32767 there) ----
    float psum = 0.0f;
#pragma unroll
    for (int k = 0; k < 32; ++k) {
        float xi = fmaxf(v[k] - rmax, lo_clip);         // exact integer difference
        float qf = floorf(xi * inv_x0i);                // q in [0,30]
        float rr = xi - x0i * qf;
        float z  = rr * (rr + b_i) + c_i;
        float ei = fmaxf(floorf(ldexpf(z, 30 - (int)qf)), 0.0f);   // exact 2^(30-q)
        float e2 = fminf(fmaxf(rintf(ei * s2), -32768.0f), 32767.0f);
        v[k] = e2;
        psum += e2;
    }

    // ---- row sum on the matrix pipe: V_WMMA_F32_16X16X4_F32 with B = ones ----
    // A(16x4): lanes0-15 {K0=p,K1=0}, lanes16-31 {K2=p,K3=0} => D[m][n] = p[m]+p[m+16]
    v2f A; A[0] = psum; A[1] = 0.0f;
    v2f B; B[0] = 1.0f; B[1] = 1.0f;
    v8f C = {};
    v8f D = __builtin_amdgcn_wmma_f32_16x16x4_f32(false, A, false, B, (short)0, C, false, false);
    float t = ((D[0] + D[1]) + (D[2] + D[3])) + ((D[4] + D[5]) + (D[6] + D[7]));
    float total = t + __shfl_xor(t, 16, 32);            // all 32 partials summed

    // ---- integer normalize + int8 output scale ----
    // fo = factor * 2^-24 is an exact power-of-two scaling of factor, so
    // floor(e2*fo) == floor((e2*factor)/2^24) bit-for-bit.
    float factor = floorf(4294967296.0f / total);       // floor(2^32 / sum)
    float fo     = factor * (1.0f / 16777216.0f);
#pragma unroll
    for (int k = 0; k < 8; ++k) {
        v4f o;
        o.x = floorf(v[4*k+0] * fo) * 0.00390625f;
        o.y = floorf(v[4*k+1] * fo) * 0.00390625f;
        o.z = floorf(v[4*k+2] * fo) * 0.00390625f;
        o.w = floorf(v[4*k+3] * fo) * 0.00390625f;
        // NT store: don't let the 100 MB output evict the L2-resident input
        __builtin_nontemporal_store(o, &orow4[k * 32 + lane]);
    }

    if (row == 0 && threadIdx.x == 0)
        out[(size_t)n_rows * 1024] = 0.00390625f;       // out_scale = 1/2^8
}

extern "C" void kernel_launch(void* const* d_in, const int* in_sizes, int n_in,
                              void* d_out, int out_size, void* d_ws, size_t ws_size,
                              hipStream_t stream) {
    const float* x = (const float*)d_in[0];
    float* out = (float*)d_out;
    unsigned int* absbits = (unsigned int*)d_ws;

    const int n      = in_sizes[0];      // 2*12*1024*1024
    const int n4     = n >> 2;
    const int n_rows = n >> 10;          // rows of 1024

    init_ws_kernel<<<1, 32, 0, stream>>>(absbits);
    absmax_kernel<<<2048, 256, 0, stream>>>(x, absbits, n4);
    isoftmax_kernel<<<(n_rows + 7) / 8, 256, 0, stream>>>(x, out, absbits, n_rows);
}